// QNetwork_72258529788550
// MI455X (gfx1250) — compile-verified
//
#include <hip/hip_runtime.h>
#include <math.h>

#define BB 16
#define SS 512
#define S2 64
#define HH 128
#define LL 3

typedef __attribute__((ext_vector_type(2))) float v2f;
typedef __attribute__((ext_vector_type(8))) float v8f;

#define WMMA_F32(A, Bv, C) \
    __builtin_amdgcn_wmma_f32_16x16x4_f32(false, (A), false, (Bv), (short)0, (C), false, false)

// Software-pipelined accumulation over one contiguous K segment.
// Per k-group: 1 B(weight) fragment + 4 A fragments feed 4 WMMAs; the next
// group's loads are issued before the current group's WMMAs (double buffer).
__device__ __forceinline__ void seg_xwt(const float*& wp, const float* xp, size_t rstep,
                                        int kcount,
                                        v8f& acc0, v8f& acc1, v8f& acc2, v8f& acc3) {
    v2f bv = *(const v2f*)wp;
    v2f a0 = *(const v2f*)(xp);
    v2f a1 = *(const v2f*)(xp + rstep);
    v2f a2 = *(const v2f*)(xp + 2 * rstep);
    v2f a3 = *(const v2f*)(xp + 3 * rstep);
    xp += 4; wp += 4;
    for (int k = 4; k < kcount; k += 4) {
        const v2f nb = *(const v2f*)wp;
        const v2f n0 = *(const v2f*)(xp);
        const v2f n1 = *(const v2f*)(xp + rstep);
        const v2f n2 = *(const v2f*)(xp + 2 * rstep);
        const v2f n3 = *(const v2f*)(xp + 3 * rstep);
        xp += 4; wp += 4;
        acc0 = WMMA_F32(a0, bv, acc0);
        acc1 = WMMA_F32(a1, bv, acc1);
        acc2 = WMMA_F32(a2, bv, acc2);
        acc3 = WMMA_F32(a3, bv, acc3);
        bv = nb; a0 = n0; a1 = n1; a2 = n2; a3 = n3;
    }
    acc0 = WMMA_F32(a0, bv, acc0);
    acc1 = WMMA_F32(a1, bv, acc1);
    acc2 = WMMA_F32(a2, bv, acc2);
    acc3 = WMMA_F32(a3, bv, acc3);
}

// Epilogue for one 16x16 tile (no pointer arrays -> no scratch risk).
// modes: 0 plain, 1 sigmoid, 2 sigmoid*E1, 3 (1-E1)*E2 + E1*tanh(v)
__device__ __forceinline__ void store_tile(v8f ac, int mbase, int cn, float bb,
                                           const float* __restrict__ E1,
                                           const float* __restrict__ E2, int ldE,
                                           float* __restrict__ Y, int ldy, int mode) {
#pragma unroll
    for (int j = 0; j < 8; ++j) {
        const int m = mbase + j;
        const float v = ac[j] + bb;
        float y;
        if (mode == 0) {
            y = v;
        } else if (mode == 1) {
            y = 1.f / (1.f + __expf(-v));
        } else if (mode == 2) {
            y = (1.f / (1.f + __expf(-v))) * E1[(size_t)m * ldE + cn];
        } else {
            const float zz = E1[(size_t)m * ldE + cn];
            const float hh = E2[(size_t)m * ldE + cn];
            y = (1.f - zz) * hh + zz * tanhf(v);
        }
        Y[(size_t)m * ldy + cn] = y;
    }
}

// ---------------------------------------------------------------------------
// Generic GEMM:  Y[m,n] = epilogue( sum_k X[m,k] * W[n,k] + bias[n] )
// Each wave computes a 64x16 slab (4 accumulators sharing one weight frag).
// Split-A handled as two sequential pipelined segments.
// ---------------------------------------------------------------------------
__global__ void gemm_xwt(const float* __restrict__ X, const float* __restrict__ X2,
                         int Ksplit, int ldx, int ldx2,
                         const float* __restrict__ W, int ldw,
                         const float* __restrict__ bias,
                         const float* __restrict__ E1, const float* __restrict__ E2,
                         int ldE,
                         float* __restrict__ Y, int ldy,
                         int N, int K, int mode) {
    const int lane = threadIdx.x & 31;
    const int wave = threadIdx.x >> 5;
    const int row0 = blockIdx.x * 64;
    const int col0 = (blockIdx.y * 4 + wave) * 16;
    if (col0 >= N) return;

    const int mr  = lane & 15;        // A row / B col within tile
    const int kh2 = (lane >> 4) * 2;  // k sub-offset

    v8f acc0 = {0.f,0.f,0.f,0.f,0.f,0.f,0.f,0.f};
    v8f acc1 = acc0, acc2 = acc0, acc3 = acc0;

    const float* wp = W + (size_t)(col0 + mr) * ldw + kh2;
    const int kend1 = (Ksplit < K) ? Ksplit : K;

    seg_xwt(wp, X + (size_t)(row0 + mr) * ldx + kh2, (size_t)16 * ldx,
            kend1, acc0, acc1, acc2, acc3);
    if (kend1 < K)
        seg_xwt(wp, X2 + (size_t)(row0 + mr) * ldx2 + kh2, (size_t)16 * ldx2,
                K - kend1, acc0, acc1, acc2, acc3);

    const int cn = col0 + (lane & 15);
    const int cmo = (lane >> 4) * 8;
    const float bb = bias ? bias[cn] : 0.f;
    store_tile(acc0, row0 + cmo,      cn, bb, E1, E2, ldE, Y, ldy, mode);
    store_tile(acc1, row0 + 16 + cmo, cn, bb, E1, E2, ldE, Y, ldy, mode);
    store_tile(acc2, row0 + 32 + cmo, cn, bb, E1, E2, ldE, Y, ldy, mode);
    store_tile(acc3, row0 + 48 + cmo, cn, bb, E1, E2, ldE, Y, ldy, mode);
}

// ---------------------------------------------------------------------------
// Batched GEMM:  Y[b] = X[b] (MxK row major) @ Bm[b] (KxN row major)
// Same 64x16 slab per wave; B fragment = two strided scalar loads shared by
// 4 WMMAs; double-buffered like seg_xwt.
// ---------------------------------------------------------------------------
__global__ void gemm_bat(const float* __restrict__ X, long long sX, int ldx,
                         const float* __restrict__ Bm, long long sB, int ldb,
                         float* __restrict__ Y, long long sY, int ldy,
                         int N, int K) {
    const int b = blockIdx.z;
    const float* Xb = X + (size_t)b * sX;
    const float* Bb = Bm + (size_t)b * sB;
    float* Yb = Y + (size_t)b * sY;

    const int lane = threadIdx.x & 31;
    const int wave = threadIdx.x >> 5;
    const int row0 = blockIdx.x * 64;
    const int col0 = (blockIdx.y * 4 + wave) * 16;
    if (col0 >= N) return;

    const int mr  = lane & 15;
    const int kh2 = (lane >> 4) * 2;

    v8f acc0 = {0.f,0.f,0.f,0.f,0.f,0.f,0.f,0.f};
    v8f acc1 = acc0, acc2 = acc0, acc3 = acc0;

    const float* xp = Xb + (size_t)(row0 + mr) * ldx + kh2;
    const float* bp = Bb + (size_t)kh2 * ldb + col0 + mr;
    const size_t rstep = (size_t)16 * ldx;
    const size_t bstep = (size_t)4 * ldb;

    v2f bv; bv.x = bp[0]; bv.y = bp[ldb];
    v2f a0 = *(const v2f*)(xp);
    v2f a1 = *(const v2f*)(xp + rstep);
    v2f a2 = *(const v2f*)(xp + 2 * rstep);
    v2f a3 = *(const v2f*)(xp + 3 * rstep);
    xp += 4; bp += bstep;

    for (int k = 4; k < K; k += 4) {
        v2f nb; nb.x = bp[0]; nb.y = bp[ldb];
        const v2f n0 = *(const v2f*)(xp);
        const v2f n1 = *(const v2f*)(xp + rstep);
        const v2f n2 = *(const v2f*)(xp + 2 * rstep);
        const v2f n3 = *(const v2f*)(xp + 3 * rstep);
        xp += 4; bp += bstep;
        acc0 = WMMA_F32(a0, bv, acc0);
        acc1 = WMMA_F32(a1, bv, acc1);
        acc2 = WMMA_F32(a2, bv, acc2);
        acc3 = WMMA_F32(a3, bv, acc3);
        bv = nb; a0 = n0; a1 = n1; a2 = n2; a3 = n3;
    }
    acc0 = WMMA_F32(a0, bv, acc0);
    acc1 = WMMA_F32(a1, bv, acc1);
    acc2 = WMMA_F32(a2, bv, acc2);
    acc3 = WMMA_F32(a3, bv, acc3);

    const int cn = col0 + (lane & 15);
    const int cmo = (lane >> 4) * 8;
#pragma unroll
    for (int j = 0; j < 8; ++j) Yb[(size_t)(row0 + cmo + j) * ldy + cn]      = acc0[j];
#pragma unroll
    for (int j = 0; j < 8; ++j) Yb[(size_t)(row0 + 16 + cmo + j) * ldy + cn] = acc1[j];
#pragma unroll
    for (int j = 0; j < 8; ++j) Yb[(size_t)(row0 + 32 + cmo + j) * ldy + cn] = acc2[j];
#pragma unroll
    for (int j = 0; j < 8; ++j) Yb[(size_t)(row0 + 48 + cmo + j) * ldy + cn] = acc3[j];
}

// ---------------------------------------------------------------------------
// Fused MLP pairwise attention:
// scores[b,q,k] = relu( sum_h relu(qp[b,q,h]+kp[b,k,h]+ba1[h]) * Wa2[h] + ba2 )
// A = mask>0 ? exp(scores) : 0 ;  attn = A / max(sum_k A, 2e-15)
// ---------------------------------------------------------------------------
__global__ void attn_kernel(const float* __restrict__ qp, const float* __restrict__ kp,
                            const float* __restrict__ ba1, const float* __restrict__ Wa2,
                            const float* __restrict__ ba2,
                            const float* __restrict__ actions_mask,
                            const float* __restrict__ state_mask,
                            float* __restrict__ attn) {
    __shared__ float Abuf[SS];
    __shared__ float wsum[8];

    const int bq = blockIdx.x;
    const int b = bq >> 6;
    const int q = bq & 63;
    const int tid = threadIdx.x;
    const int wave = tid >> 5;
    const int lane = tid & 31;

    const float4 qv = *(const float4*)(qp + (size_t)bq * HH + lane * 4);
    const float4 bav = *(const float4*)(ba1 + lane * 4);
    const float4 wav = *(const float4*)(Wa2 + lane * 4);
    const float am = actions_mask[b * S2 + q];
    const float b2 = ba2[0];

    float lsum = 0.f;
    for (int k = wave; k < SS; k += 8) {
        const float4 kv = *(const float4*)(kp + ((size_t)b * SS + k) * HH + lane * 4);
        float ps = fmaxf(qv.x + kv.x + bav.x, 0.f) * wav.x
                 + fmaxf(qv.y + kv.y + bav.y, 0.f) * wav.y
                 + fmaxf(qv.z + kv.z + bav.z, 0.f) * wav.z
                 + fmaxf(qv.w + kv.w + bav.w, 0.f) * wav.w;
#pragma unroll
        for (int off = 16; off > 0; off >>= 1)
            ps += __shfl_xor(ps, off, 32);
        if (lane == 0) {
            const float sc = fmaxf(ps + b2, 0.f);
            const float mk = am * state_mask[b * SS + k];
            const float e = (mk > 0.f) ? __expf(sc) : 0.f;
            Abuf[k] = e;
            lsum += e;
        }
    }
    if (lane == 0) wsum[wave] = lsum;
    __syncthreads();

    float denom = 0.f;
#pragma unroll
    for (int i = 0; i < 8; ++i) denom += wsum[i];
    denom = fmaxf(denom, 2e-15f);

    for (int k = tid; k < SS; k += 256)
        attn[(size_t)bq * SS + k] = Abuf[k] / denom;
}

// sem/val head: sem[b,h] = sum_q sf[b,q,h] / sum_q mask ; val[b] = sem.Wv + bv
__global__ void sem_val_kernel(const float* __restrict__ sf,
                               const float* __restrict__ actions_mask,
                               const float* __restrict__ Wv, const float* __restrict__ bv,
                               float* __restrict__ val) {
    __shared__ float red[HH];
    const int b = blockIdx.x;
    const int h = threadIdx.x;
    float s = 0.f;
    for (int q = 0; q < S2; ++q) s += sf[((size_t)b * S2 + q) * HH + h];
    float num = 0.f;
    for (int q = 0; q < S2; ++q) num += actions_mask[b * S2 + q];
    red[h] = (s / num) * Wv[h];
    __syncthreads();
    for (int st = 64; st > 0; st >>= 1) {
        if (h < st) red[h] += red[h + st];
        __syncthreads();
    }
    if (h == 0) val[b] = red[0] + bv[0];
}

// bilinear dueling head: adv[bs,l] = actions[bs,:] . (weight[l] @ sf[bs,:]) + bias[l]
__global__ void adv_kernel(const float* __restrict__ actions, const float* __restrict__ sf,
                           const float* __restrict__ weight, const float* __restrict__ bias,
                           float* __restrict__ adv) {
    __shared__ float f[HH];
    __shared__ float red[HH];
    const int bs = blockIdx.x;
    const int h = threadIdx.x;
    f[h] = sf[(size_t)bs * HH + h];
    __syncthreads();
    const float ah = actions[(size_t)bs * HH + h];
    for (int l = 0; l < LL; ++l) {
        const float* wl = weight + (size_t)l * HH * HH + (size_t)h * HH;
        float t = 0.f;
#pragma unroll 4
        for (int k = 0; k < HH; ++k) t += wl[k] * f[k];
        red[h] = ah * t;
        __syncthreads();
        for (int st = 64; st > 0; st >>= 1) {
            if (h < st) red[h] += red[h + st];
            __syncthreads();
        }
        if (h == 0) adv[bs * LL + l] = red[0] + bias[l];
        __syncthreads();
    }
}

// q = val[b] + adv - mean_b(adv)
__global__ void final_kernel(const float* __restrict__ adv, const float* __restrict__ val,
                             float* __restrict__ out) {
    __shared__ float red[256];
    const int b = blockIdx.x;
    const int t = threadIdx.x;
    const int n = S2 * LL; // 192
    const float a = (t < n) ? adv[b * n + t] : 0.f;
    red[t] = a;
    __syncthreads();
    for (int st = 128; st > 0; st >>= 1) {
        if (t < st) red[t] += red[t + st];
        __syncthreads();
    }
    const float mean = red[0] / (float)n;
    if (t < n) out[b * n + t] = val[b] + a - mean;
}

// ---------------------------------------------------------------------------
extern "C" void kernel_launch(void* const* d_in, const int* in_sizes, int n_in,
                              void* d_out, int out_size, void* d_ws, size_t ws_size,
                              hipStream_t stream) {
    const float* states       = (const float*)d_in[0];
    const float* state_mask   = (const float*)d_in[1];
    const float* actions      = (const float*)d_in[2];
    const float* actions_mask = (const float*)d_in[3];
    const float* adj          = (const float*)d_in[4];
    const float* Wn  = (const float*)d_in[5];
    const float* bn  = (const float*)d_in[6];
    const float* Wu  = (const float*)d_in[7];
    const float* bu  = (const float*)d_in[8];
    const float* Wr  = (const float*)d_in[9];
    const float* br  = (const float*)d_in[10];
    const float* Wt  = (const float*)d_in[11];
    const float* bt  = (const float*)d_in[12];
    const float* Wa1 = (const float*)d_in[13];
    const float* ba1 = (const float*)d_in[14];
    const float* Wa2 = (const float*)d_in[15];
    const float* ba2 = (const float*)d_in[16];
    const float* Wv  = (const float*)d_in[17];
    const float* bv  = (const float*)d_in[18];
    const float* weight = (const float*)d_in[19];
    const float* bias   = (const float*)d_in[20];

    float* ws = (float*)d_ws;
    const size_t BSH = (size_t)BB * SS * HH;      // 1,048,576
    float* h0 = ws;
    float* h1 = h0 + BSH;
    float* p  = h1 + BSH;
    float* a  = p  + BSH;                         // B*S*2H
    float* rh = a  + 2 * BSH;
    float* z  = rh + BSH;
    float* qp = z  + BSH;                         // B*S2*H
    float* kp = qp + (size_t)BB * S2 * HH;
    float* at = kp + BSH;                         // B*S2*S
    float* sf = at + (size_t)BB * S2 * SS;        // B*S2*H
    float* vl = sf + (size_t)BB * S2 * HH;        // B
    float* ad = vl + 64;                          // B*S2*L

    const int M1 = BB * SS;    // 8192
    const dim3 blk(128);
    const dim3 gM1(M1 / 64, 2);          // 128 x 2
    const dim3 gAdj(SS / 64, 2, BB);     // 8 x 2 x 16
    const dim3 gQP(BB * S2 / 64, 2);     // 16 x 2
    const dim3 gSF(S2 / 64, 2, BB);      // 1 x 2 x 16

    const float* hcur = states;
    float* hbuf[2] = {h0, h1};
    const int KBIG = 1 << 30;

    for (int step = 0; step < 3; ++step) {
        // p = h @ Wn^T + bn
        gemm_xwt<<<gM1, blk, 0, stream>>>(hcur, nullptr, KBIG, HH, HH,
                                          Wn, HH, bn, nullptr, nullptr, HH,
                                          p, HH, HH, HH, 0);
        // a[:, :H] = adj @ p ; a[:, H:] = adj @ h   (batched over B)
        gemm_bat<<<gAdj, blk, 0, stream>>>(adj, (long long)SS * SS, SS,
                                           p, (long long)SS * HH, HH,
                                           a, (long long)SS * 2 * HH, 2 * HH,
                                           HH, SS);
        gemm_bat<<<gAdj, blk, 0, stream>>>(adj, (long long)SS * SS, SS,
                                           hcur, (long long)SS * HH, HH,
                                           a + HH, (long long)SS * 2 * HH, 2 * HH,
                                           HH, SS);
        // z = sigmoid(a @ Wu^T + bu)
        gemm_xwt<<<gM1, blk, 0, stream>>>(a, nullptr, KBIG, 2 * HH, 2 * HH,
                                          Wu, 2 * HH, bu, nullptr, nullptr, HH,
                                          z, HH, HH, 2 * HH, 1);
        // rh = sigmoid(a @ Wr^T + br) * h
        gemm_xwt<<<gM1, blk, 0, stream>>>(a, nullptr, KBIG, 2 * HH, 2 * HH,
                                          Wr, 2 * HH, br, hcur, nullptr, HH,
                                          rh, HH, HH, 2 * HH, 2);
        // hnew = (1-z)*h + z*tanh(concat(p, rh) @ Wt^T + bt)
        float* hnew = hbuf[step & 1];
        gemm_xwt<<<gM1, blk, 0, stream>>>(p, rh, HH, HH, HH,
                                          Wt, 2 * HH, bt, z, hcur, HH,
                                          hnew, HH, HH, 2 * HH, 3);
        hcur = hnew;
    }
    const float* gout = hcur;  // [B,S,H]

    // qp = actions @ Wa1[:, :H]^T ; kp = out @ Wa1[:, H:]^T
    gemm_xwt<<<gQP, blk, 0, stream>>>(actions, nullptr, KBIG, HH, HH,
                                      Wa1, 2 * HH, nullptr, nullptr, nullptr, HH,
                                      qp, HH, HH, HH, 0);
    gemm_xwt<<<gM1, blk, 0, stream>>>(gout, nullptr, KBIG, HH, HH,
                                      Wa1 + HH, 2 * HH, nullptr, nullptr, nullptr, HH,
                                      kp, HH, HH, HH, 0);

    // fused scores + masked exp + normalize -> attn
    attn_kernel<<<dim3(BB * S2), dim3(256), 0, stream>>>(qp, kp, ba1, Wa2, ba2,
                                                         actions_mask, state_mask, at);

    // states_feat = attn @ out  (batched)
    gemm_bat<<<gSF, blk, 0, stream>>>(at, (long long)S2 * SS, SS,
                                      gout, (long long)SS * HH, HH,
                                      sf, (long long)S2 * HH, HH,
                                      HH, SS);

    // heads
    sem_val_kernel<<<dim3(BB), dim3(HH), 0, stream>>>(sf, actions_mask, Wv, bv, vl);
    adv_kernel<<<dim3(BB * S2), dim3(HH), 0, stream>>>(actions, sf, weight, bias, ad);
    final_kernel<<<dim3(BB), dim3(256), 0, stream>>>(ad, vl, (float*)d_out);
}